// ESNReservoir_37890201486023
// MI455X (gfx1250) — compile-verified
//
#include <hip/hip_runtime.h>
#include <math.h>

// Problem constants (match reference)
#define RES      4096   // reservoir size N
#define TSTEPS   4096   // time steps T
#define IOCH     32     // input/output channels
#define ALPHA    0.9f   // leaking rate

// Persistent recurrence kernel configuration
#define NB_REC   256                        // blocks (all co-resident)
#define TPB_REC  256                        // threads per block = 8 waves
#define ROWS_PER_BLOCK (RES / NB_REC)       // 16 rows of W per block
#define WAVES_PER_BLOCK (TPB_REC / 32)      // 8
#define ROWS_PER_WAVE  (ROWS_PER_BLOCK / WAVES_PER_BLOCK) // 2

// Use gfx1250 async global->LDS copies for the per-step h broadcast
#define USE_ASYNC_LDS 1

typedef __attribute__((ext_vector_type(2))) float v2f;
typedef __attribute__((ext_vector_type(8))) float v8f;

// ---------------------------------------------------------------------------
// Init: zero h0 and the grid-barrier state (runs every call -> deterministic)
// ---------------------------------------------------------------------------
__global__ void esn_init_kernel(float* __restrict__ hg, int* __restrict__ bar) {
  int i = blockIdx.x * blockDim.x + threadIdx.x;
  if (i < RES) hg[i] = 0.0f;
  if (i == 0) { bar[0] = 0; bar[1] = 0; }
}

// ---------------------------------------------------------------------------
// Device-wide sense barrier (agent scope). Writers must __threadfence()
// before entering. t = current step; sense counts completed steps.
// ---------------------------------------------------------------------------
__device__ __forceinline__ void grid_barrier(int* cnt, int* sense, int t) {
  __syncthreads();
  if (threadIdx.x == 0) {
    int prev = __hip_atomic_fetch_add(cnt, 1, __ATOMIC_ACQ_REL,
                                      __HIP_MEMORY_SCOPE_AGENT);
    if (prev == (int)gridDim.x - 1) {
      __hip_atomic_store(cnt, 0, __ATOMIC_RELAXED, __HIP_MEMORY_SCOPE_AGENT);
      __hip_atomic_fetch_add(sense, 1, __ATOMIC_RELEASE,
                             __HIP_MEMORY_SCOPE_AGENT);
    } else {
      while (__hip_atomic_load(sense, __ATOMIC_ACQUIRE,
                               __HIP_MEMORY_SCOPE_AGENT) <= t) {
        __builtin_amdgcn_s_sleep(1);  // polite spin
      }
    }
    __threadfence();
  }
  __syncthreads();
}

// ---------------------------------------------------------------------------
// Persistent recurrence: h <- (1-a)h + a*tanh(Win@x_t + W@h), S[t]=h.
// Each block owns 16 fixed rows of W (256 KB, L2-resident across all steps).
// Each wave computes 2 rows per step sharing one LDS h stream.
// ---------------------------------------------------------------------------
__global__ void __launch_bounds__(TPB_REC)
esn_recurrence_kernel(const float* __restrict__ x,     // [T, 32]
                      const float* __restrict__ Win,   // [RES, 32]
                      const float* __restrict__ W,     // [RES, RES]
                      float* __restrict__ S,           // [T, RES] out states
                      float* __restrict__ hg,          // [RES] global h
                      int* __restrict__ bar) {         // [2] cnt, sense
  __shared__ __align__(16) float hs[RES];
  __shared__ float xs[IOCH];

  const int tid  = threadIdx.x;
  const int lane = tid & 31;
  const int wave = tid >> 5;
  const int r0 = blockIdx.x * ROWS_PER_BLOCK + wave * ROWS_PER_WAVE;
  const int r1 = r0 + 1;
  int* cnt   = bar;
  int* sense = bar + 1;

  const float4* w0row = (const float4*)(W + (size_t)r0 * RES);
  const float4* w1row = (const float4*)(W + (size_t)r1 * RES);

  for (int t = 0; t < TSTEPS; ++t) {
    // ---- Stage h (16 KB) and x_t (128 B) into LDS ----
#if USE_ASYNC_LDS && defined(__gfx1250__)
    {
      // 1024 float4 / 256 threads = 4 async b128 copies per thread.
      // vdst = wave-relative LDS byte address (generic-ptr low 32 bits),
      // vaddr = 32-bit offset against the hg base held in an SGPR pair.
      const unsigned long long gbase = (unsigned long long)(uintptr_t)hg;
#pragma unroll
      for (int j = 0; j < 4; ++j) {
        const int i = tid + j * TPB_REC;                 // float4 index
        const unsigned int loff = (unsigned int)(uintptr_t)(&hs[i * 4]);
        const unsigned int goff = (unsigned int)(i * 16);
        asm volatile("global_load_async_to_lds_b128 %0, %1, %2"
                     :
                     : "v"(loff), "v"(goff), "s"(gbase)
                     : "memory");
      }
      if (tid < IOCH) xs[tid] = x[t * IOCH + tid];
      asm volatile("s_wait_asynccnt 0x0" ::: "memory");
    }
#else
    {
      const float4* hg4 = (const float4*)hg;
      float4* hs4 = (float4*)hs;
      for (int i = tid; i < RES / 4; i += TPB_REC) hs4[i] = hg4[i];
      if (tid < IOCH) xs[tid] = x[t * IOCH + tid];
    }
#endif
    __syncthreads();

    // ---- Two rows per wave, sharing one LDS h stream ----
    const float4* h4 = (const float4*)hs;
    float acc0 = Win[r0 * IOCH + lane] * xs[lane];
    float acc1 = Win[r1 * IOCH + lane] * xs[lane];
#pragma unroll 4
    for (int i = lane; i < RES / 4; i += 32) {
      const float4 h = h4[i];
      const float4 a = w0row[i];
      const float4 b = w1row[i];
      acc0 = fmaf(a.x, h.x, acc0);
      acc0 = fmaf(a.y, h.y, acc0);
      acc0 = fmaf(a.z, h.z, acc0);
      acc0 = fmaf(a.w, h.w, acc0);
      acc1 = fmaf(b.x, h.x, acc1);
      acc1 = fmaf(b.y, h.y, acc1);
      acc1 = fmaf(b.z, h.z, acc1);
      acc1 = fmaf(b.w, h.w, acc1);
    }
    // wave32 butterfly reductions
#pragma unroll
    for (int off = 16; off > 0; off >>= 1) {
      acc0 += __shfl_xor(acc0, off, 32);
      acc1 += __shfl_xor(acc1, off, 32);
    }

    if (lane == 0) {
      const float v0 = (1.0f - ALPHA) * hs[r0] + ALPHA * tanhf(acc0);
      const float v1 = (1.0f - ALPHA) * hs[r1] + ALPHA * tanhf(acc1);
      S[(size_t)t * RES + r0] = v0;
      S[(size_t)t * RES + r1] = v1;
      hg[r0] = v0;
      hg[r1] = v1;
    }
    __threadfence();           // publish h updates device-wide
    grid_barrier(cnt, sense, t);
  }
}

// ---------------------------------------------------------------------------
// Output projection: Y[T,32] = S[T,RES] @ Wout^T + bias via
// V_WMMA_F32_16X16X4_F32. One wave per 16(M)x32(N) strip: each A fragment
// feeds two WMMAs (both N tiles). 256 waves = 32 blocks x 256 threads.
// ---------------------------------------------------------------------------
__global__ void __launch_bounds__(256)
esn_output_gemm_kernel(const float* __restrict__ S,     // [T, RES]
                       const float* __restrict__ Wout,  // [32, RES]
                       const float* __restrict__ bias,  // [32]
                       float* __restrict__ Y) {         // [T, 32]
  const int lane = threadIdx.x & 31;
  const int wave = threadIdx.x >> 5;
  const int mTile = blockIdx.x * (blockDim.x >> 5) + wave;  // 0..255
  const int t0 = mTile * 16;
  const int lm = lane & 15;        // row/col within half-wave
  const int hi = lane >> 4;        // 0: K+0..1 / M 0..7 ; 1: K+2..3 / M 8..15

  // C layout: lanes 0-15 -> (M=v, N=lm); lanes 16-31 -> (M=v+8, N=lm).
  // Every (v,lane) slot is a distinct element D[m][n]: seed all with bias[n].
  const float b0 = bias[lm];        // N tile 0: channels 0..15
  const float b1 = bias[16 + lm];   // N tile 1: channels 16..31
  v8f c0, c1;
#pragma unroll
  for (int i = 0; i < 8; ++i) { c0[i] = b0; c1[i] = b1; }

  const float* arow  = S + (size_t)(t0 + lm) * RES + hi * 2;       // A 16x4
  const float* brow0 = Wout + (size_t)(lm) * RES + hi * 2;         // B k x n0
  const float* brow1 = Wout + (size_t)(16 + lm) * RES + hi * 2;    // B k x n1

  for (int kb = 0; kb < RES; kb += 4) {
    v2f a, b, b2;
    a[0]  = arow[kb + 0];   a[1]  = arow[kb + 1];
    b[0]  = brow0[kb + 0];  b[1]  = brow0[kb + 1];
    b2[0] = brow1[kb + 0];  b2[1] = brow1[kb + 1];
    c0 = __builtin_amdgcn_wmma_f32_16x16x4_f32(
        false, a, false, b,  (short)0, c0, false, false);
    c1 = __builtin_amdgcn_wmma_f32_16x16x4_f32(
        false, a, false, b2, (short)0, c1, false, false);
  }

#pragma unroll
  for (int v = 0; v < 8; ++v) {
    const int m = v + hi * 8;
    Y[(size_t)(t0 + m) * IOCH + lm]      = c0[v];
    Y[(size_t)(t0 + m) * IOCH + 16 + lm] = c1[v];
  }
}

// ---------------------------------------------------------------------------
// hT = states[T-1]  (appended after outputs in d_out)
// ---------------------------------------------------------------------------
__global__ void esn_copy_ht_kernel(const float* __restrict__ src,
                                   float* __restrict__ dst) {
  int i = blockIdx.x * blockDim.x + threadIdx.x;
  if (i < RES) dst[i] = src[i];
}

// ---------------------------------------------------------------------------
extern "C" void kernel_launch(void* const* d_in, const int* in_sizes, int n_in,
                              void* d_out, int out_size, void* d_ws, size_t ws_size,
                              hipStream_t stream) {
  const float* x    = (const float*)d_in[0];  // [1, T, 32]
  const float* Win  = (const float*)d_in[1];  // [RES, 32]
  const float* W    = (const float*)d_in[2];  // [RES, RES]
  const float* Wout = (const float*)d_in[3];  // [32, RES]
  const float* bias = (const float*)d_in[4];  // [32]
  float* out = (float*)d_out;                 // [T*32] outputs then [RES] hT

  // Workspace: states (64 MB) | h (16 KB) | barrier (2 ints)
  float* S  = (float*)d_ws;
  float* hg = S + (size_t)TSTEPS * RES;
  int*   bar = (int*)(hg + RES);

  // 1) zero h0 + barrier state (every call; replay-safe)
  esn_init_kernel<<<(RES + 255) / 256, 256, 0, stream>>>(hg, bar);

  // 2) persistent recurrence over all 4096 steps (device-wide barrier/step)
  esn_recurrence_kernel<<<NB_REC, TPB_REC, 0, stream>>>(x, Win, W, S, hg, bar);

  // 3) output projection with fp32 WMMA
  esn_output_gemm_kernel<<<32, 256, 0, stream>>>(S, Wout, bias, out);

  // 4) final hidden state
  esn_copy_ht_kernel<<<(RES + 255) / 256, 256, 0, stream>>>(
      S + (size_t)(TSTEPS - 1) * RES, out + (size_t)TSTEPS * IOCH);
}